// TransportMap_6167573037527
// MI455X (gfx1250) — compile-verified
//
#include <hip/hip_runtime.h>
#include <hip/hip_bf16.h>
#include <math.h>

typedef float v2f __attribute__((ext_vector_type(2)));
typedef float v8f __attribute__((ext_vector_type(8)));

#define NTHREADS 256
#define NROWS    128
#define DPAD_MAX 68            // max (m+p) padded: m<=64, small p, even stride -> 8B-aligned v2f LDS loads
#define ALPHA_C  2.0625f       // 1/NUG_MULT^2 + 2  (constant since nugSd = 4*nugMean)

// One workgroup per batch b. 256 threads = 8 waves; wave w owns tile-row w of the 8x8 tile grid.
__global__ void __launch_bounds__(NTHREADS)
vecchia_nll_kernel(const float* __restrict__ theta,
                   const float* __restrict__ Y,      // [128][N]
                   const float* __restrict__ X,      // [128][N][p]
                   const int*   __restrict__ NNmax,  // [N][mMax]
                   const int*   __restrict__ mPtr,   // scalar m
                   float* __restrict__ nll,          // [N] per-batch negative loglik
                   int N, int p, int mMax)
{
    extern __shared__ char smemRaw[];
    float* sS    = (float*)smemRaw;            // [128][128] gram -> G -> Cholesky L
    float* sW    = sS + NROWS * NROWS;         // [128][DPAD_MAX]
    float* sDiag = sW + NROWS * DPAD_MAX;      // [128] saved gram diagonal
    float* sZ    = sDiag + NROWS;              // [128] solve vector
    float* sYsc  = sZ + NROWS;                 // [64]  yscale
    float* sXsc  = sYsc + 64;                  // [8]   xscale
    int*   sNN   = (int*)(sXsc + 8);           // [64]  neighbor indices

    const int b   = blockIdx.x;
    const int tid = threadIdx.x;

    int m = *mPtr; if (m > mMax) m = mMax; if (m > 64) m = 64;
    const int d       = m + p;
    const int dpad    = (d + 3) & ~3;
    const int kkSteps = dpad >> 2;

    // ---- per-batch scalars (uniform across block) ----
    const float t0 = theta[0], t1 = theta[1], t2 = theta[2];
    const float t3 = theta[3], t4 = theta[4], t5 = theta[5];
    const float logscal = -logf((float)b + 1.0f);
    float nugMean = expf(logscal * t1 + t0) - 1e-5f;
    nugMean = (nugMean > 0.0f ? nugMean : 0.0f) + 1e-5f;
    const float beta    = nugMean * (ALPHA_C - 1.0f);
    const float sigma   = expf(logscal * t4 + t3);
    const float sig2    = sigma * sigma;
    const float lenScal = expf(t5) * sqrtf(3.0f);       // exp(th5)*sqrt(2*SMOOTH)
    const float invLen2 = 1.0f / (lenScal * lenScal);
    const float c3      = sqrtf(3.0f);
    const int   ncol    = (b < m) ? b : m;

    // ---- small tables ----
    if (tid < m) {
        sNN[tid]  = NNmax[b * mMax + tid];
        sYsc[tid] = sqrtf(expf((float)(tid + 1) * t2));
    }
    if (tid < p) sXsc[tid] = expf(theta[3 + tid] + logscal * theta[3 + p + tid]);
    __syncthreads();

    // ---- build W (gather Y columns, append scaled X), zero-pad to DPAD_MAX ----
    for (int idx = tid; idx < NROWS * DPAD_MAX; idx += NTHREADS) {
        const int i = idx / DPAD_MAX;
        const int j = idx - i * DPAD_MAX;
        float v = 0.0f;
        if (j < ncol)              v = Y[i * N + sNN[j]] * sYsc[j];
        else if (j >= m && j < d)  v = X[(i * N + b) * p + (j - m)] * sXsc[j - m];
        sW[idx] = v;
    }
    __syncthreads();

    // ---- gram S = W * W^T via fp32 WMMA 16x16x4 ----
    const int lane  = tid & 31;
    const int wave  = tid >> 5;
    const int rA    = lane & 15;            // M (for A) / N-source-row (for B=W^T)
    const int kh    = (lane >> 4) << 1;     // K half offset: 0 or 2
    const int ti    = wave;                 // tile row (8 waves -> 8 tile rows)
    const int mbase = ti * 16 + ((lane >> 4) << 3);
    const int cn    = lane & 15;

    for (int tj = 0; tj < 8; ++tj) {
        v8f acc = {0.f, 0.f, 0.f, 0.f, 0.f, 0.f, 0.f, 0.f};
        for (int kk = 0; kk < kkSteps; ++kk) {
            const int kc = kk * 4 + kh;
            v2f a  = *(const v2f*)&sW[(ti * 16 + rA) * DPAD_MAX + kc];
            v2f bb = *(const v2f*)&sW[(tj * 16 + rA) * DPAD_MAX + kc]; // B = W^T tile
            acc = __builtin_amdgcn_wmma_f32_16x16x4_f32(
                      false, a, false, bb, (short)0, acc, false, false);
        }
#pragma unroll
        for (int v = 0; v < 8; ++v)
            sS[(mbase + v) * NROWS + tj * 16 + cn] = acc[v];
    }
    __syncthreads();

    // ---- save gram diagonal (needed for d2 after in-place overwrite) ----
    if (tid < NROWS) sDiag[tid] = sS[tid * NROWS + tid];
    __syncthreads();

    // ---- G = (lin + sig2*matern(d))/nugMean + I   (G=I for b==0) ----
    for (int idx = tid; idx < NROWS * NROWS; idx += NTHREADS) {
        const int i = idx >> 7;
        const int j = idx & 127;
        float g;
        if (b == 0) {
            g = (i == j) ? 1.0f : 0.0f;
        } else {
            const float lin = sS[idx];
            float d2 = (sDiag[i] + sDiag[j] - 2.0f * lin) * invLen2;
            d2 = fmaxf(d2, 1e-12f);
            const float dd  = sqrtf(d2);
            const float mat = (1.0f + c3 * dd) * expf(-c3 * dd);
            g = (lin + sig2 * mat) / nugMean + ((i == j) ? 1.0f : 0.0f);
        }
        sS[idx] = g;
    }
    __syncthreads();

    // ---- in-place Cholesky (lower), right-looking; thread t owns row t ----
    for (int k = 0; k < NROWS; ++k) {
        if (tid == 0) sS[k * NROWS + k] = sqrtf(sS[k * NROWS + k]);
        __syncthreads();
        const float dinv = 1.0f / sS[k * NROWS + k];
        if (tid > k && tid < NROWS) sS[tid * NROWS + k] *= dinv;
        __syncthreads();
        if (tid > k && tid < NROWS) {
            const float lik = sS[tid * NROWS + k];
            for (int j = k + 1; j <= tid; ++j)
                sS[tid * NROWS + j] -= lik * sS[j * NROWS + k];
        }
        __syncthreads();
    }

    // ---- forward substitution: L z = Y[:, b] ----
    if (tid < NROWS) sZ[tid] = Y[tid * N + b];
    __syncthreads();
    for (int j = 0; j < NROWS; ++j) {
        if (tid == j) sZ[j] /= sS[j * NROWS + j];
        __syncthreads();
        if (tid > j && tid < NROWS) sZ[tid] -= sS[tid * NROWS + j] * sZ[j];
        __syncthreads();
    }

    // ---- scalar tail ----
    if (tid == 0) {
        float ss = 0.0f, logdet = 0.0f;
        for (int i = 0; i < NROWS; ++i) {
            ss     += sZ[i] * sZ[i];
            logdet += logf(sS[i * NROWS + i]);
        }
        const float alphaPost = ALPHA_C + (float)NROWS * 0.5f;
        const float betaPost  = beta + 0.5f * ss;
        const float loglik = -logdet + ALPHA_C * logf(beta)
                           - alphaPost * logf(betaPost)
                           + lgammaf(alphaPost) - lgammaf(ALPHA_C);
        nll[b] = -loglik;
    }
}

// Deterministic fixed-order reduction of per-batch NLL.
__global__ void __launch_bounds__(NTHREADS)
sum_kernel(const float* __restrict__ nll, float* __restrict__ out, int N)
{
    __shared__ float red[NTHREADS];
    float s = 0.0f;
    for (int i = threadIdx.x; i < N; i += NTHREADS) s += nll[i];
    red[threadIdx.x] = s;
    __syncthreads();
    for (int off = NTHREADS / 2; off > 0; off >>= 1) {
        if (threadIdx.x < off) red[threadIdx.x] += red[threadIdx.x + off];
        __syncthreads();
    }
    if (threadIdx.x == 0) out[0] = red[0];
}

extern "C" void kernel_launch(void* const* d_in, const int* in_sizes, int n_in,
                              void* d_out, int out_size, void* d_ws, size_t ws_size,
                              hipStream_t stream)
{
    const float* theta = (const float*)d_in[0];
    const float* Y     = (const float*)d_in[1];
    const float* X     = (const float*)d_in[2];
    const int*   NN    = (const int*)d_in[3];
    const int*   mPtr  = (const int*)d_in[4];

    const int n    = 128;                       // fixed by reference
    const int N    = in_sizes[1] / n;           // 2048
    const int p    = in_sizes[2] / in_sizes[1]; // 1
    const int mMax = in_sizes[3] / N;           // 64

    float* nll = (float*)d_ws;                  // N floats of scratch

    const size_t shmem =
        (size_t)(NROWS * NROWS + NROWS * DPAD_MAX + NROWS + NROWS + 64 + 8) * sizeof(float)
        + 64 * sizeof(int);                     // ~99.5 KB of the 320 KB WGP LDS

    vecchia_nll_kernel<<<N, NTHREADS, shmem, stream>>>(theta, Y, X, NN, mPtr,
                                                       nll, N, p, mMax);
    sum_kernel<<<1, NTHREADS, 0, stream>>>(nll, (float*)d_out, N);
}